// RGCN_867583393980
// MI455X (gfx1250) — compile-verified
//
#include <hip/hip_runtime.h>
#include <hip/hip_bf16.h>

// ---------------------------------------------------------------------------
// RGCN (2-layer, 8 relations, scatter-MEAN) for MI455X / gfx1250.
// Aggregate-then-transform: scatter-mean raw 128-ch features per relation into
// a [N,128] slab, then dense V_WMMA_F32_16X16X4_F32 GEMMs accumulate into the
// output. Root linears ride the same WMMA kernel (bias preloaded into C; node
// tiles are type-homogeneous since N0 % 16 == 0).
//
// Round-1 fixes (from gfx1250 disasm): B fragments are now batched 32-deep
// into registers per N-tile (one s_wait_loadcnt per 32 WMMAs instead of one
// per WMMA), and C_OUT / root-mode are template parameters so full tiles have
// no OOB masking VALU ops in front of the WMMA chain. 3125 M-tiles = 5 waves
// x 625 blocks exactly -> no guards, EXEC all-ones at every WMMA.
// ---------------------------------------------------------------------------

#define RG_N_NODES 50000
#define RG_N0      30000
#define RG_N1      20000
#define RG_N_EDGES 600000
#define RG_NUM_ET  8
#define RG_IN_C    128
#define RG_HID_C   128
#define RG_OUT_C   40
#define RG_KD      128                 // inner dim of every GEMM
#define RG_M_TILES (RG_N_NODES / 16)   // 3125 = 5 * 625, exact

typedef __attribute__((ext_vector_type(2))) float v2f;
typedef __attribute__((ext_vector_type(8))) float v8f;

// ---------------------------------------------------------------------------
// x = select(node_type==0, x0[local_idx], emb1[local_idx])   [N, 128]
// ---------------------------------------------------------------------------
__global__ void rg_build_x(const float* __restrict__ x0,
                           const float* __restrict__ emb1,
                           const int*   __restrict__ node_type,
                           const int*   __restrict__ lidx,
                           float* __restrict__ x)
{
    int gid = blockIdx.x * blockDim.x + threadIdx.x;
    int n   = gid >> 5;
    int c4  = gid & 31;
    if (n >= RG_N_NODES) return;
    int ty = node_type[n];
    int l  = lidx[n];
    const float4* sp;
    if (ty == 0) {
        int ll = l < 0 ? 0 : (l > RG_N0 - 1 ? RG_N0 - 1 : l);
        sp = (const float4*)(x0 + (size_t)ll * RG_IN_C);
    } else {
        int ll = l < 0 ? 0 : (l > RG_N1 - 1 ? RG_N1 - 1 : l);
        sp = (const float4*)(emb1 + (size_t)ll * RG_IN_C);
    }
    ((float4*)(x + (size_t)n * RG_IN_C))[c4] = sp[c4];
}

__global__ void rg_zero_f32x4(float4* __restrict__ p, int n4)
{
    int i = blockIdx.x * blockDim.x + threadIdx.x;
    if (i < n4) p[i] = make_float4(0.f, 0.f, 0.f, 0.f);
}

// Edge counts per (type, dst): cnt[t*N + dst] += 1  (fp32 atomics)
__global__ void rg_count_edges(const int* __restrict__ ei,
                               const int* __restrict__ et,
                               float* __restrict__ cnt)
{
    int e = blockIdx.x * blockDim.x + threadIdx.x;
    if (e >= RG_N_EDGES) return;
    int t   = et[e];
    int dst = ei[RG_N_EDGES + e];
    atomicAdd(cnt + (size_t)t * RG_N_NODES + dst, 1.0f);
}

// Scatter-mean for one relation t: agg[dst, :] += X[src, :] / max(cnt[t,dst],1)
// 8 threads per edge, 16 channels each.
__global__ void rg_scatter_mean(const float* __restrict__ X,
                                const int*   __restrict__ ei,
                                const int*   __restrict__ et,
                                const float* __restrict__ cnt,
                                float* __restrict__ agg,
                                int t)
{
    int gid  = blockIdx.x * blockDim.x + threadIdx.x;
    int e    = gid >> 3;
    int part = gid & 7;
    if (e >= RG_N_EDGES) return;
    if (et[e] != t) return;
    int dst = ei[RG_N_EDGES + e];
    int src = ei[e];
    float inv = 1.0f / fmaxf(cnt[(size_t)t * RG_N_NODES + dst], 1.0f);
    const float* xs = X + (size_t)src * RG_KD + part * 16;
    float*       ad = agg + (size_t)dst * RG_KD + part * 16;
#pragma unroll
    for (int c = 0; c < 16; ++c)
        atomicAdd(ad + c, xs[c] * inv);
}

// ---------------------------------------------------------------------------
// WMMA f32 16x16x4 GEMM: out[Mtile, :C_OUT] (+)= A[Mtile, 0:128] @ W.T
//   ROOT=true : W = [2, C_OUT, 128] selected by tile node type, C seeded with
//               bias -> overwrite out.
//   ROOT=false: C loaded from out, W = one relation's [C_OUT, 128] -> add.
// One wave per 16-row M tile (5 waves x 625 blocks = 3125 tiles exact).
// Per N-tile: 32 B fragments batched into regs (loads pipeline 32-deep),
// then a back-to-back 32-WMMA accumulation chain. Tail tile (C_OUT=40) uses
// compile-time-known clamped/masked loads; EXEC stays all-ones at every WMMA.
// ---------------------------------------------------------------------------
template <int C_OUT, bool ROOT>
__global__ __launch_bounds__(160)
void rg_gemm_wmma(const float* __restrict__ A,
                  const float* __restrict__ W,
                  const float* __restrict__ bias,
                  const int*   __restrict__ node_type,
                  float* __restrict__ out)
{
    constexpr int NT_FULL  = C_OUT / 16;
    constexpr int TAIL_N0  = NT_FULL * 16;
    constexpr bool HAS_TAIL = (C_OUT % 16) != 0;

    const int wave = threadIdx.x >> 5;
    const int lane = threadIdx.x & 31;
    const int mt   = blockIdx.x * 5 + wave;   // exact cover, no guard
    const int m0   = mt * 16;
    const int row  = m0 + (lane & 15);
    const int kofs = (lane >> 4) << 1;        // lanes 0-15: K=0,1 ; 16-31: K=2,3
    const int rofs = (lane >> 4) << 3;        // C/D row offset (+8 for hi lanes)
    const int ncol = lane & 15;

    // Preload the full A strip: A[row, 4*kk + kofs .. +1], kk = 0..31
    v2f a[32];
    {
        const float* Ap = A + (size_t)row * RG_KD + kofs;
#pragma unroll
        for (int kk = 0; kk < 32; ++kk) {
            a[kk].x = Ap[4 * kk + 0];
            a[kk].y = Ap[4 * kk + 1];
        }
    }

    const float* Wp = W;
    const float* bp = bias;
    if (ROOT) {
        int ty = node_type[m0];               // tile is type-homogeneous
        Wp += (size_t)ty * C_OUT * RG_KD;
        bp += (size_t)ty * C_OUT;
    }

    // ---- full 16-wide N tiles: no masking anywhere ----
    for (int nt = 0; nt < NT_FULL; ++nt) {
        const int col = nt * 16 + ncol;
        const float* Wrow = Wp + (size_t)col * RG_KD + kofs;

        v2f b[32];                            // batch all B fragments first
#pragma unroll
        for (int kk = 0; kk < 32; ++kk) {
            b[kk].x = Wrow[4 * kk + 0];
            b[kk].y = Wrow[4 * kk + 1];
        }

        v8f c;
        if (ROOT) {
            float bv = bp[col];
#pragma unroll
            for (int i = 0; i < 8; ++i) c[i] = bv;
        } else {
#pragma unroll
            for (int i = 0; i < 8; ++i)
                c[i] = out[(size_t)(m0 + i + rofs) * C_OUT + col];
        }

#pragma unroll
        for (int kk = 0; kk < 32; ++kk)
            c = __builtin_amdgcn_wmma_f32_16x16x4_f32(
                    false, a[kk], false, b[kk], (short)0, c, false, false);

#pragma unroll
        for (int i = 0; i < 8; ++i)
            out[(size_t)(m0 + i + rofs) * C_OUT + col] = c[i];
    }

    // ---- tail tile (only instantiated for C_OUT=40) ----
    if (HAS_TAIL) {
        const int   col  = TAIL_N0 + ncol;
        const int   colc = col < C_OUT ? col : (C_OUT - 1);
        const float msk  = col < C_OUT ? 1.0f : 0.0f;
        const float* Wrow = Wp + (size_t)colc * RG_KD + kofs;

        v2f b[32];
#pragma unroll
        for (int kk = 0; kk < 32; ++kk) {
            b[kk].x = msk * Wrow[4 * kk + 0];
            b[kk].y = msk * Wrow[4 * kk + 1];
        }

        v8f c;
        if (ROOT) {
            float bv = msk * bp[colc];
#pragma unroll
            for (int i = 0; i < 8; ++i) c[i] = bv;
        } else {
#pragma unroll
            for (int i = 0; i < 8; ++i)
                c[i] = msk * out[(size_t)(m0 + i + rofs) * C_OUT + colc];
        }

#pragma unroll
        for (int kk = 0; kk < 32; ++kk)
            c = __builtin_amdgcn_wmma_f32_16x16x4_f32(
                    false, a[kk], false, b[kk], (short)0, c, false, false);

#pragma unroll
        for (int i = 0; i < 8; ++i)
            if (col < C_OUT)
                out[(size_t)(m0 + i + rofs) * C_OUT + col] = c[i];
    }
}

__global__ void rg_relu_f32x4(float4* __restrict__ p, int n4)
{
    int i = blockIdx.x * blockDim.x + threadIdx.x;
    if (i >= n4) return;
    float4 v = p[i];
    v.x = fmaxf(v.x, 0.f); v.y = fmaxf(v.y, 0.f);
    v.z = fmaxf(v.z, 0.f); v.w = fmaxf(v.w, 0.f);
    p[i] = v;
}

// In-place log_softmax over the 40-channel rows of out. One thread per node.
__global__ void rg_log_softmax(float* __restrict__ out)
{
    int n = blockIdx.x * blockDim.x + threadIdx.x;
    if (n >= RG_N_NODES) return;
    float* row = out + (size_t)n * RG_OUT_C;
    float m = -3.402823466e38f;
#pragma unroll
    for (int c = 0; c < RG_OUT_C; ++c) m = fmaxf(m, row[c]);
    float s = 0.f;
#pragma unroll
    for (int c = 0; c < RG_OUT_C; ++c) s += expf(row[c] - m);
    float ls = logf(s) + m;
#pragma unroll
    for (int c = 0; c < RG_OUT_C; ++c) row[c] = row[c] - ls;
}

// ---------------------------------------------------------------------------
extern "C" void kernel_launch(void* const* d_in, const int* in_sizes, int n_in,
                              void* d_out, int out_size, void* d_ws, size_t ws_size,
                              hipStream_t stream)
{
    const float* x0      = (const float*)d_in[0];
    const float* emb1    = (const float*)d_in[1];
    const float* rel_w1  = (const float*)d_in[2];   // [8,128,128]
    const float* root_w1 = (const float*)d_in[3];   // [2,128,128]
    const float* root_b1 = (const float*)d_in[4];   // [2,128]
    const float* rel_w2  = (const float*)d_in[5];   // [8,40,128]
    const float* root_w2 = (const float*)d_in[6];   // [2,40,128]
    const float* root_b2 = (const float*)d_in[7];   // [2,40]
    const int*   ei      = (const int*)d_in[8];     // [2,600000]
    const int*   et      = (const int*)d_in[9];     // [600000]
    const int*   nty     = (const int*)d_in[10];    // [50000]
    const int*   lidx    = (const int*)d_in[11];    // [50000]
    float*       out     = (float*)d_out;           // [50000,40]

    // Workspace (bytes): x 25.6MB | h 25.6MB | agg slab 25.6MB | cnt 1.6MB
    char*  ws  = (char*)d_ws;
    float* x   = (float*)(ws);
    float* h   = (float*)(ws + (size_t)25600000);
    float* agg = (float*)(ws + (size_t)51200000);
    float* cnt = (float*)(ws + (size_t)76800000);

    const int TB = 256;
    const dim3 gGemm(RG_M_TILES / 5);   // 625 blocks x 5 waves = 3125 tiles
    const dim3 bGemm(160);

    // x = grouped input
    rg_build_x<<<(RG_N_NODES * 32 + TB - 1) / TB, TB, 0, stream>>>(x0, emb1, nty, lidx, x);

    // edge counts per (type,dst) — shared by both layers
    rg_zero_f32x4<<<((RG_NUM_ET * RG_N_NODES / 4) + TB - 1) / TB, TB, 0, stream>>>(
        (float4*)cnt, RG_NUM_ET * RG_N_NODES / 4);
    rg_count_edges<<<(RG_N_EDGES + TB - 1) / TB, TB, 0, stream>>>(ei, et, cnt);

    // ---------------- Layer 1: x[N,128] -> h[N,128] ----------------
    rg_gemm_wmma<RG_HID_C, true><<<gGemm, bGemm, 0, stream>>>(x, root_w1, root_b1, nty, h);
    for (int t = 0; t < RG_NUM_ET; ++t) {
        rg_zero_f32x4<<<((RG_N_NODES * RG_KD / 4) + TB - 1) / TB, TB, 0, stream>>>(
            (float4*)agg, RG_N_NODES * RG_KD / 4);
        rg_scatter_mean<<<(RG_N_EDGES * 8 + TB - 1) / TB, TB, 0, stream>>>(
            x, ei, et, cnt, agg, t);
        rg_gemm_wmma<RG_HID_C, false><<<gGemm, bGemm, 0, stream>>>(
            agg, rel_w1 + (size_t)t * RG_HID_C * RG_KD, nullptr, nty, h);
    }
    rg_relu_f32x4<<<((RG_N_NODES * RG_HID_C / 4) + TB - 1) / TB, TB, 0, stream>>>(
        (float4*)h, RG_N_NODES * RG_HID_C / 4);

    // ---------------- Layer 2: h[N,128] -> out[N,40] ----------------
    rg_gemm_wmma<RG_OUT_C, true><<<gGemm, bGemm, 0, stream>>>(h, root_w2, root_b2, nty, out);
    for (int t = 0; t < RG_NUM_ET; ++t) {
        rg_zero_f32x4<<<((RG_N_NODES * RG_KD / 4) + TB - 1) / TB, TB, 0, stream>>>(
            (float4*)agg, RG_N_NODES * RG_KD / 4);
        rg_scatter_mean<<<(RG_N_EDGES * 8 + TB - 1) / TB, TB, 0, stream>>>(
            h, ei, et, cnt, agg, t);
        rg_gemm_wmma<RG_OUT_C, false><<<gGemm, bGemm, 0, stream>>>(
            agg, rel_w2 + (size_t)t * RG_OUT_C * RG_KD, nullptr, nty, out);
    }
    rg_log_softmax<<<(RG_N_NODES + TB - 1) / TB, TB, 0, stream>>>(out);
}